// GIN_1803886264475
// MI455X (gfx1250) — compile-verified
//
#include <hip/hip_runtime.h>
#include <stdint.h>

typedef float v2f __attribute__((ext_vector_type(2)));
typedef float v8f __attribute__((ext_vector_type(8)));

#define N_NODES 100000
#define N_EDGES 1600000
#define IN_C    64
#define HID_C   128
#define OUT_C   64

// Guaranteed-hardware fp32 atomic add (global_atomic_add_f32, not a CAS loop).
__device__ __forceinline__ void atomic_add_f32(float* p, float v) {
  unsafeAtomicAdd(p, v);
}

// ---------------------------------------------------------------------------
// agg = x  (vectorized copy; initializes the "(1+eps)*x_i" self term, eps=0)
// ---------------------------------------------------------------------------
__global__ void gin_copy_f4(const float4* __restrict__ src,
                            float4* __restrict__ dst, int n4) {
  int i = blockIdx.x * blockDim.x + threadIdx.x;
  if (i < n4) dst[i] = src[i];
}

// ---------------------------------------------------------------------------
// agg[dst[e]] += x[src[e]]   (C channels, 4 channels per thread, f32 atomics)
// ---------------------------------------------------------------------------
template <int C>
__global__ void gin_scatter_add(const float* __restrict__ x,
                                const long long* __restrict__ src,
                                const long long* __restrict__ dst,
                                float* __restrict__ agg) {
  const int TPE = C / 4;  // threads per edge
  long long gid = (long long)blockIdx.x * blockDim.x + threadIdx.x;
  int e = (int)(gid / TPE);
  if (e >= N_EDGES) return;
  int c = (int)(gid % TPE) * 4;
  int s = (int)src[e];
  int d = (int)dst[e];
  const float4 v = *(const float4*)(x + (long long)s * C + c);
  float* p = agg + (long long)d * C + c;
  atomic_add_f32(p + 0, v.x);
  atomic_add_f32(p + 1, v.y);
  atomic_add_f32(p + 2, v.z);
  atomic_add_f32(p + 3, v.w);
}

// ---------------------------------------------------------------------------
// out[M, NCOLS] = act(A[M, K] @ W[K, NCOLS] + bias)
// One wave computes one 16x16 tile via V_WMMA_F32_16X16X4_F32, K in steps of 4.
// grid.x = M/16 tiles; blockDim.x = (NCOLS/16) * 32 (one wave per N-tile).
//
// A layout (16x4 f32): all lanes hold row m = lane&15; lanes 0-15 carry K pair
//   {0,1}, lanes 16-31 carry {2,3} in regs {v0,v1}. B (4x16) is symmetric:
//   lane holds column n = lane&15, same K-pair split.
// C/D layout (16x16 f32, 8 regs): lane n = lane&15; reg r holds row
//   r + (lane>>4)*8.
// ---------------------------------------------------------------------------
template <int K, int NCOLS, bool RELU>
__global__ void gin_gemm_wmma(const float* __restrict__ A,
                              const float* __restrict__ W,
                              const float* __restrict__ bias,
                              float* __restrict__ out) {
  const int wave  = threadIdx.x >> 5;
  const int lane  = threadIdx.x & 31;
  const int lo    = lane & 15;
  const int hi    = lane >> 4;   // 0 or 1
  const int kb    = hi * 2;      // K offset within a 4-wide step
  const int mbase = blockIdx.x * 16;
  const int nbase = wave * 16;

  v8f acc = {0.f, 0.f, 0.f, 0.f, 0.f, 0.f, 0.f, 0.f};

  const float* arow = A + (long long)(mbase + lo) * K;  // row m = lane&15
  const float* wcol = W + nbase + lo;                   // column n = lane&15

#pragma unroll
  for (int k0 = 0; k0 < K; k0 += 4) {
    v2f a = *(const v2f*)(arow + k0 + kb);  // A[m][k0+kb], A[m][k0+kb+1]
    v2f b;
    b.x = wcol[(long long)(k0 + kb) * NCOLS];        // W[k0+kb  ][n]
    b.y = wcol[(long long)(k0 + kb + 1) * NCOLS];    // W[k0+kb+1][n]
    acc = __builtin_amdgcn_wmma_f32_16x16x4_f32(
        /*neg_a=*/false, a, /*neg_b=*/false, b,
        /*c_mod=*/(short)0, acc, /*reuse_a=*/false, /*reuse_b=*/false);
  }

  const float bv = bias[nbase + lo];           // per-column bias
  const int mrow = mbase + hi * 8;
#pragma unroll
  for (int r = 0; r < 8; ++r) {
    float v = acc[r] + bv;
    if (RELU) v = fmaxf(v, 0.0f);
    out[(long long)(mrow + r) * NCOLS + nbase + lo] = v;
  }
}

// ---------------------------------------------------------------------------
extern "C" void kernel_launch(void* const* d_in, const int* in_sizes, int n_in,
                              void* d_out, int out_size, void* d_ws,
                              size_t ws_size, hipStream_t stream) {
  const float* node_emb    = (const float*)d_in[0];       // [100000, 64]
  const float* W1          = (const float*)d_in[1];       // [64, 128]
  const float* b1          = (const float*)d_in[2];       // [128]
  const float* W2          = (const float*)d_in[3];       // [128, 64]
  const float* b2          = (const float*)d_in[4];       // [64]
  const long long* edge    = (const long long*)d_in[5];   // [2, 1600000] int64
  const long long* src     = edge;
  const long long* dst     = edge + N_EDGES;

  char* ws = (char*)d_ws;
  float* agg1 = (float*)(ws);                                  // 25.6 MB
  float* h    = (float*)(ws + (size_t)N_NODES * IN_C * 4);     // 51.2 MB
  float* agg2 = (float*)(ws + (size_t)N_NODES * (IN_C + HID_C) * 4);  // 51.2 MB
  float* out  = (float*)d_out;                                 // [100000, 64]

  // ---- layer 1: agg1 = x + scatter_sum(x[src] -> dst) ----
  {
    int n4 = N_NODES * IN_C / 4;
    gin_copy_f4<<<(n4 + 255) / 256, 256, 0, stream>>>(
        (const float4*)node_emb, (float4*)agg1, n4);
  }
  {
    long long tot = (long long)N_EDGES * (IN_C / 4);
    gin_scatter_add<IN_C><<<(int)((tot + 255) / 256), 256, 0, stream>>>(
        node_emb, src, dst, agg1);
  }
  // h = relu(agg1 @ W1 + b1)
  gin_gemm_wmma<IN_C, HID_C, true>
      <<<N_NODES / 16, (HID_C / 16) * 32, 0, stream>>>(agg1, W1, b1, h);

  // ---- layer 2: agg2 = h + scatter_sum(h[src] -> dst) ----
  {
    int n4 = N_NODES * HID_C / 4;
    gin_copy_f4<<<(n4 + 255) / 256, 256, 0, stream>>>(
        (const float4*)h, (float4*)agg2, n4);
  }
  {
    long long tot = (long long)N_EDGES * (HID_C / 4);
    gin_scatter_add<HID_C><<<(int)((tot + 255) / 256), 256, 0, stream>>>(
        h, src, dst, agg2);
  }
  // out = agg2 @ W2 + b2
  gin_gemm_wmma<HID_C, OUT_C, false>
      <<<N_NODES / 16, (OUT_C / 16) * 32, 0, stream>>>(agg2, W2, b2, out);
}